// DL_model_1735166788200
// MI455X (gfx1250) — compile-verified
//
#include <hip/hip_runtime.h>
#include <math.h>

#define B_ 32
#define N_ 512
#define L_ 2048
#define D_ 128
#define INF_ (-50000000000.0f)
#define NEGBIG (-3.0e38f)

typedef __attribute__((ext_vector_type(2))) float v2f;
typedef __attribute__((ext_vector_type(8))) float v8f;

// ---------------------------------------------------------------------------
// K1: sim[b,n,l] = sum_d xd[b,n,d]*xt[b,d,l], masked to INF_ on padding.
// Wave32 WMMA f32 16x16x4. Each wave computes a 16x64 strip (4 accumulators,
// A-fragment reused across the 4 column tiles). Block = 128 threads = 4 waves
// covering 256 columns. Grid = (L/256, N/16, B).
// ---------------------------------------------------------------------------
__global__ __launch_bounds__(128) void k_gemm_mask(
    const float* __restrict__ xd, const float* __restrict__ xt,
    const int* __restrict__ nobjs, const int* __restrict__ slens,
    float* __restrict__ sim)
{
    const int b    = blockIdx.z;
    const int n0   = blockIdx.y << 4;
    const int wave = threadIdx.x >> 5;
    const int lane = threadIdx.x & 31;
    const int half = lane >> 4;       // lanes 0-15 vs 16-31
    const int lid  = lane & 15;
    const int l0   = (blockIdx.x << 8) + (wave << 6);

    // A: 16x4 fp32 fragment. lane m (half 0) holds K = k..k+1, half 1 holds K = k+2..k+3
    const float* __restrict__ Ap =
        xd + ((size_t)b * N_ + (n0 + lid)) * D_ + 2 * half;
    // B: 4x16 fp32 fragment. VGPR j holds row K = k + 2*half + j, col = l
    const float* __restrict__ Bp =
        xt + (size_t)b * D_ * L_ + (size_t)(2 * half) * L_ + l0 + lid;

    v8f acc0 = {}, acc1 = {}, acc2 = {}, acc3 = {};
    #pragma unroll 4
    for (int k = 0; k < D_; k += 4) {
        v2f a;
        a.x = Ap[k];
        a.y = Ap[k + 1];
        const float* bp = Bp + (size_t)k * L_;
        v2f bt;
        bt.x = bp[0];       bt.y = bp[L_ + 0];
        acc0 = __builtin_amdgcn_wmma_f32_16x16x4_f32(false, a, false, bt, (short)0, acc0, false, false);
        bt.x = bp[16];      bt.y = bp[L_ + 16];
        acc1 = __builtin_amdgcn_wmma_f32_16x16x4_f32(false, a, false, bt, (short)0, acc1, false, false);
        bt.x = bp[32];      bt.y = bp[L_ + 32];
        acc2 = __builtin_amdgcn_wmma_f32_16x16x4_f32(false, a, false, bt, (short)0, acc2, false, false);
        bt.x = bp[48];      bt.y = bp[L_ + 48];
        acc3 = __builtin_amdgcn_wmma_f32_16x16x4_f32(false, a, false, bt, (short)0, acc3, false, false);
    }

    const int nObj = nobjs[b];
    const int sl   = slens[b];
    const bool lb0 = (l0 + lid      ) >= sl;
    const bool lb1 = (l0 + lid + 16) >= sl;
    const bool lb2 = (l0 + lid + 32) >= sl;
    const bool lb3 = (l0 + lid + 48) >= sl;
    float* __restrict__ sb = sim + (size_t)b * N_ * L_;

    // D layout: VGPR i <-> row M = i (lanes 0-15) / i+8 (lanes 16-31)
    #pragma unroll
    for (int i = 0; i < 8; ++i) {
        const int row = n0 + i + 8 * half;
        const bool rbad = row >= nObj;
        const size_t base = (size_t)row * L_ + l0 + lid;
        sb[base +  0] = (rbad || lb0) ? INF_ : acc0[i];
        sb[base + 16] = (rbad || lb1) ? INF_ : acc1[i];
        sb[base + 32] = (rbad || lb2) ? INF_ : acc2[i];
        sb[base + 48] = (rbad || lb3) ? INF_ : acc3[i];
    }
}

// ---------------------------------------------------------------------------
// K2: per-row (b,n) online softmax stats over L. Grid (N, B), block 256.
// ---------------------------------------------------------------------------
__global__ __launch_bounds__(256) void k_rowstats(
    const float* __restrict__ sim,
    float* __restrict__ rowmax, float* __restrict__ rowsum)
{
    const int b = blockIdx.y, n = blockIdx.x, tid = threadIdx.x;
    const float* __restrict__ row = sim + ((size_t)b * N_ + n) * L_;
    float m = NEGBIG, s = 0.f;
    for (int l = tid; l < L_; l += 256) {
        const float x = row[l];
        const float M = fmaxf(m, x);
        s = s * __expf(m - M) + __expf(x - M);
        m = M;
    }
    __shared__ float sm[256], ss[256];
    sm[tid] = m; ss[tid] = s;
    __syncthreads();
    for (int off = 128; off > 0; off >>= 1) {
        if (tid < off) {
            const float m1 = sm[tid], s1 = ss[tid];
            const float m2 = sm[tid + off], s2 = ss[tid + off];
            const float M = fmaxf(m1, m2);
            sm[tid] = M;
            ss[tid] = s1 * __expf(m1 - M) + s2 * __expf(m2 - M);
        }
        __syncthreads();
    }
    if (tid == 0) { rowmax[b * N_ + n] = sm[0]; rowsum[b * N_ + n] = ss[0]; }
}

// ---------------------------------------------------------------------------
// K3: per-column (b,l) online softmax stats over N. Grid (L/64, B), block (64,4).
// Coalesced: consecutive tx -> consecutive l.
// ---------------------------------------------------------------------------
__global__ __launch_bounds__(256) void k_colstats(
    const float* __restrict__ sim,
    float* __restrict__ colmax, float* __restrict__ colsum)
{
    const int b  = blockIdx.y;
    const int tx = threadIdx.x, ty = threadIdx.y;
    const int l  = (blockIdx.x << 6) + tx;
    const float* __restrict__ base = sim + (size_t)b * N_ * L_ + l;
    float m = NEGBIG, s = 0.f;
    for (int n = ty; n < N_; n += 4) {
        const float x = base[(size_t)n * L_];
        const float M = fmaxf(m, x);
        s = s * __expf(m - M) + __expf(x - M);
        m = M;
    }
    __shared__ float sm[4][64], ss[4][64];
    sm[ty][tx] = m; ss[ty][tx] = s;
    __syncthreads();
    if (ty == 0) {
        #pragma unroll
        for (int j = 1; j < 4; ++j) {
            const float m2 = sm[j][tx], s2 = ss[j][tx];
            const float M = fmaxf(m, m2);
            s = s * __expf(m - M) + s2 * __expf(m2 - M);
            m = M;
        }
        colmax[b * L_ + l] = m; colsum[b * L_ + l] = s;
    }
}

// ---------------------------------------------------------------------------
// K4: normalize -> write s_a (softmax over N, uses column stats) and
// a_s (softmax over L, uses row stats); fuse rowAgg[b,n] = sum_l s_a.
// Grid (N, B), block 256.
// ---------------------------------------------------------------------------
__global__ __launch_bounds__(256) void k_normalize(
    const float* __restrict__ sim,
    const float* __restrict__ rowmax, const float* __restrict__ rowsum,
    const float* __restrict__ colmax, const float* __restrict__ colsum,
    float* __restrict__ out_sa, float* __restrict__ out_as,
    float* __restrict__ rowAgg)
{
    const int b = blockIdx.y, n = blockIdx.x, tid = threadIdx.x;
    const size_t roff = ((size_t)b * N_ + n) * L_;
    const float rm   = rowmax[b * N_ + n];
    const float rinv = 1.f / rowsum[b * N_ + n];
    const float* __restrict__ cm = colmax + (size_t)b * L_;
    const float* __restrict__ cs = colsum + (size_t)b * L_;
    float agg = 0.f;
    for (int l = tid; l < L_; l += 256) {
        const float x  = sim[roff + l];
        const float as = __expf(x - rm) * rinv;
        const float sa = __expf(x - cm[l]) / cs[l];
        out_as[roff + l] = as;
        out_sa[roff + l] = sa;
        agg += sa;
    }
    __shared__ float red[256];
    red[tid] = agg;
    __syncthreads();
    for (int off = 128; off > 0; off >>= 1) {
        if (tid < off) red[tid] += red[tid + off];
        __syncthreads();
    }
    if (tid == 0) rowAgg[b * N_ + n] = red[0];
}

// ---------------------------------------------------------------------------
// K5: colAgg[b,l] = sum_n a_s[b,n,l]. Grid (L/64, B), block (64,4).
// ---------------------------------------------------------------------------
__global__ __launch_bounds__(256) void k_colagg(
    const float* __restrict__ a_s, float* __restrict__ colAgg)
{
    const int b  = blockIdx.y;
    const int tx = threadIdx.x, ty = threadIdx.y;
    const int l  = (blockIdx.x << 6) + tx;
    const float* __restrict__ base = a_s + (size_t)b * N_ * L_ + l;
    float s = 0.f;
    for (int n = ty; n < N_; n += 4) s += base[(size_t)n * L_];
    __shared__ float ss[4][64];
    ss[ty][tx] = s;
    __syncthreads();
    if (ty == 0) {
        s += ss[1][tx] + ss[2][tx] + ss[3][tx];
        colAgg[b * L_ + l] = s;
    }
}

// ---------------------------------------------------------------------------
// K6: feature1[b,d] = sum_n xd[b,n,d]*rowAgg[b,n]
//     feature2[b,d] = sum_l xt[b,d,l]*colAgg[b,l]
// Grid B, block D (=128). Tiny.
// ---------------------------------------------------------------------------
__global__ __launch_bounds__(128) void k_features(
    const float* __restrict__ xd, const float* __restrict__ xt,
    const float* __restrict__ rowAgg, const float* __restrict__ colAgg,
    float* __restrict__ feat)
{
    const int b = blockIdx.x, d = threadIdx.x;
    const float* __restrict__ xdb = xd + (size_t)b * N_ * D_ + d;
    const float* __restrict__ ra  = rowAgg + (size_t)b * N_;
    float f1 = 0.f;
    for (int n = 0; n < N_; ++n) f1 += xdb[(size_t)n * D_] * ra[n];
    const float* __restrict__ xtb = xt + ((size_t)b * D_ + d) * L_;
    const float* __restrict__ ca  = colAgg + (size_t)b * L_;
    float f2 = 0.f;
    for (int l = 0; l < L_; ++l) f2 += xtb[l] * ca[l];
    feat[b * 2 * D_ + d]      = f1;
    feat[b * 2 * D_ + D_ + d] = f2;
}

// ---------------------------------------------------------------------------
extern "C" void kernel_launch(void* const* d_in, const int* in_sizes, int n_in,
                              void* d_out, int out_size, void* d_ws, size_t ws_size,
                              hipStream_t stream)
{
    const float* xd    = (const float*)d_in[0];
    const float* xt    = (const float*)d_in[1];
    const int*   nobjs = (const int*)d_in[2];
    const int*   slens = (const int*)d_in[3];

    const size_t BNL = (size_t)B_ * N_ * L_;

    // workspace layout (floats): sim | rowmax | rowsum | colmax | colsum | rowAgg | colAgg
    float* sim    = (float*)d_ws;
    float* rowmax = sim + BNL;
    float* rowsum = rowmax + (size_t)B_ * N_;
    float* colmax = rowsum + (size_t)B_ * N_;
    float* colsum = colmax + (size_t)B_ * L_;
    float* rowAgg = colsum + (size_t)B_ * L_;
    float* colAgg = rowAgg + (size_t)B_ * N_;

    // output layout (return order): feature_cat [B,2D] | s_a [B,N,L] | a_s [B,N,L]
    float* feat   = (float*)d_out;
    float* out_sa = feat + (size_t)B_ * 2 * D_;
    float* out_as = out_sa + BNL;

    k_gemm_mask<<<dim3(L_ / 256, N_ / 16, B_), 128, 0, stream>>>(xd, xt, nobjs, slens, sim);
    k_rowstats<<<dim3(N_, B_), 256, 0, stream>>>(sim, rowmax, rowsum);
    k_colstats<<<dim3(L_ / 64, B_), dim3(64, 4), 0, stream>>>(sim, colmax, colsum);
    k_normalize<<<dim3(N_, B_), 256, 0, stream>>>(sim, rowmax, rowsum, colmax, colsum,
                                                  out_sa, out_as, rowAgg);
    k_colagg<<<dim3(L_ / 64, B_), dim3(64, 4), 0, stream>>>(out_as, colAgg);
    k_features<<<B_, D_, 0, stream>>>(xd, xt, rowAgg, colAgg, feat);
}